// TransformedReLU_26998164423204
// MI455X (gfx1250) — compile-verified
//
#include <hip/hip_runtime.h>
#include <hip/hip_bf16.h>

// Problem geometry (fixed by the reference): E=64 zonotope rows, content = 8*32*32 = 8192.
#define E_ROWS 64
#define N_ELEM 8192
#define NV     (N_ELEM / 4)   // 2048 float4 per row

typedef float v2f __attribute__((ext_vector_type(2)));
typedef float v4f __attribute__((ext_vector_type(4)));   // native vector: OK for nontemporal builtin
typedef float v8f __attribute__((ext_vector_type(8)));

// ---------------------------------------------------------------------------
// Kernel 1: err[j] = sum_{i=1..63} |x[i][j]| via V_WMMA_F32_16X16X4_F32,
// then head row / lambda / delta / body scale factor / cross flags.
//
// WMMA mapping: D(16x16) = A(16x4, all ones) * B(4x16 of |x|) accumulated over
// 16 K-chunks => D[0][n] = sum over 64 rows of |x[row][col_base+n]|.
// B layout: column n = lane%16; lanes 0-15 supply K = 4t+{0,1} in {b.x,b.y},
// lanes 16-31 supply K = 4t+{2,3}. (With A == ones the sum is invariant to
// any K-slot permutation, so only the column-per-lane property matters.)
// One wave handles 16 columns; 512 waves cover all 8192 columns.
// ---------------------------------------------------------------------------
__global__ void zono_relu_stats(const float* __restrict__ x,
                                float* __restrict__ out_head,  // out row 0
                                float* __restrict__ f,         // body scale per column
                                float* __restrict__ delta2,    // delta/2 per column
                                int*   __restrict__ cross_flag)
{
    const int lane  = threadIdx.x & 31;
    const int wave  = (blockIdx.x * blockDim.x + threadIdx.x) >> 5;
    const int col   = wave * 16 + (lane & 15);
    const int khalf = (lane >> 4) ? 2 : 0;   // upper half-wave holds K = 2,3

    v2f a; a.x = 1.0f; a.y = 1.0f;           // ones A-matrix (16x4)
    v8f acc = {};

#pragma unroll
    for (int t = 0; t < 16; ++t) {           // 16 chunks of K=4 -> 64 rows
        const int r0 = 4 * t + khalf;
        v2f b;
        b.x = fabsf(x[(size_t)r0 * N_ELEM + col]);
        b.y = fabsf(x[(size_t)(r0 + 1) * N_ELEM + col]);
        // 8 args: (neg_a, A, neg_b, B, c_mod, C, reuse_a, reuse_b)
        acc = __builtin_amdgcn_wmma_f32_16x16x4_f32(
                  false, a, false, b, (short)0, acc, false, false);
    }

    // C/D layout: lanes 0-15, VGPR0 => D[M=0][N=lane] = full column abs-sum.
    if (lane < 16) {
        const float x0    = x[col];
        const float err   = acc[0] - fabsf(x0);   // exclude center row
        const float upper = x0 + err;
        const float lower = x0 - err;
        const float cross  = (lower * upper < 0.0f) ? 1.0f : 0.0f;
        const float nonneg = (lower >= 0.0f) ? 1.0f : 0.0f;
        float lam = nonneg + cross * (upper / (upper - lower));
        if (isnan(lam)) lam = 0.5f;               // reference's NaN fixup
        const float delta = fmaxf(-lam * lower, (1.0f - lam) * upper);

        out_head[col]   = (0.5f * delta + lam * x0) * cross + x0 * nonneg;
        f[col]          = lam * cross + nonneg;
        delta2[col]     = 0.5f * delta;
        cross_flag[col] = (cross != 0.0f) ? 1 : 0;
    }
}

// ---------------------------------------------------------------------------
// Kernel 2: rows[k] = clamp(cumsum(cross)[k] - 1, 0, N-1).
// Single 256-thread block; 32 elements/thread + Hillis-Steele block scan.
// ---------------------------------------------------------------------------
__global__ void zono_scan_rows(const int* __restrict__ cross_flag,
                               int* __restrict__ rows)
{
    __shared__ int partial[256];
    const int t    = threadIdx.x;
    const int base = t * 32;

    int s = 0;
#pragma unroll
    for (int j = 0; j < 32; ++j) s += cross_flag[base + j];

    partial[t] = s;
    __syncthreads();
#pragma unroll
    for (int off = 1; off < 256; off <<= 1) {
        const int mine  = partial[t];
        const int other = (t >= off) ? partial[t - off] : 0;
        __syncthreads();
        partial[t] = mine + other;
        __syncthreads();
    }
    int run = partial[t] - s;                 // exclusive prefix for this thread

#pragma unroll
    for (int j = 0; j < 32; ++j) {
        run += cross_flag[base + j];
        int r = run - 1;
        r = (r < 0) ? 0 : ((r > N_ELEM - 1) ? (N_ELEM - 1) : r);
        rows[base + j] = r;
    }
}

// ---------------------------------------------------------------------------
// Kernel 3: body rows 1..63: out[i][j] = x[i][j] * f[j]  (128-bit vectors, 2 MB)
// ---------------------------------------------------------------------------
__global__ void zono_body(const v4f* __restrict__ x4,
                          const v4f* __restrict__ f4,
                          v4f* __restrict__ out4)
{
    const int idx = blockIdx.x * blockDim.x + threadIdx.x;   // 0 .. 63*NV-1
    if (idx >= 63 * NV) return;
    const int row = idx / NV;                // 0..62  -> out row row+1
    const int cv  = idx - row * NV;
    const v4f xv = x4[(size_t)(row + 1) * NV + cv];
    const v4f fv = f4[cv];
    out4[(size_t)(row + 1) * NV + cv] = xv * fv;   // elementwise on ext vectors
}

// ---------------------------------------------------------------------------
// Kernel 4: non-temporal 128-bit zero-fill of the 8192x8192 tail (256 MiB).
// This is the bandwidth-bound stage: ~11.5 us floor at 23.3 TB/s. NT stores
// keep the 270 MB stream from thrashing the 192 MB L2.
// ---------------------------------------------------------------------------
__global__ void zono_tail_zero(v4f* __restrict__ tail4, long n4)
{
    const long stride = (long)gridDim.x * blockDim.x;
    const v4f z = {0.0f, 0.0f, 0.0f, 0.0f};
    for (long i = (long)blockIdx.x * blockDim.x + threadIdx.x; i < n4; i += stride)
        __builtin_nontemporal_store(z, &tail4[i]);
}

// ---------------------------------------------------------------------------
// Kernel 5: sparse scatter of delta/2 into the zeroed tail.
// ---------------------------------------------------------------------------
__global__ void zono_scatter(const int* __restrict__ cross_flag,
                             const int* __restrict__ rows,
                             const float* __restrict__ delta2,
                             float* __restrict__ tail)
{
    const int k = blockIdx.x * blockDim.x + threadIdx.x;
    if (k < N_ELEM && cross_flag[k])
        tail[(size_t)rows[k] * N_ELEM + k] = delta2[k];
}

// ---------------------------------------------------------------------------
extern "C" void kernel_launch(void* const* d_in, const int* in_sizes, int n_in,
                              void* d_out, int out_size, void* d_ws, size_t ws_size,
                              hipStream_t stream)
{
    (void)in_sizes; (void)n_in; (void)out_size; (void)ws_size;

    const float* x   = (const float*)d_in[0];          // (64, 8192) f32
    float*       out = (float*)d_out;                  // (1+63+8192, 8192) f32

    // Workspace layout: f | delta2 | cross | rows  (4 * 8192 * 4B = 128 KB)
    float* f          = (float*)d_ws;
    float* delta2     = f + N_ELEM;
    int*   cross_flag = (int*)(delta2 + N_ELEM);
    int*   rows       = cross_flag + N_ELEM;

    // 1) WMMA abs-column-sum + per-column stats. 512 waves = 16384 threads.
    zono_relu_stats<<<64, 256, 0, stream>>>(x, out, f, delta2, cross_flag);

    // 2) compaction row ids (single-block scan over 8192 flags).
    zono_scan_rows<<<1, 256, 0, stream>>>(cross_flag, rows);

    // 3) scaled error-term body rows (63 * 8192 floats).
    zono_body<<<(63 * NV + 255) / 256, 256, 0, stream>>>(
        (const v4f*)x, (const v4f*)f, (v4f*)out);

    // 4) + 5) tail: zero-fill then scatter.
    float* tail = out + (size_t)E_ROWS * N_ELEM;       // row 64 onward
    const long n4 = (long)N_ELEM * N_ELEM / 4;         // 16,777,216 x 128-bit
    zono_tail_zero<<<4096, 256, 0, stream>>>((v4f*)tail, n4);
    zono_scatter<<<N_ELEM / 256, 256, 0, stream>>>(cross_flag, rows, delta2, tail);
}